// RnnGmm_47115791237746
// MI455X (gfx1250) — compile-verified
//
#include <hip/hip_runtime.h>
#include <hip/hip_bf16.h>
#include <math.h>

// ---- problem constants (match reference) ----
#define DIM   128
#define HID   512
#define NC    128
#define NBAT  512
#define IN_W  (2*DIM + 1)   // 257
#define MLP_H (4*HID)       // 2048

typedef __attribute__((ext_vector_type(2))) float v2f;
typedef __attribute__((ext_vector_type(8))) float v8f;

// ---- GEMM tiling ----
#define BM   64    // batch-rows per block
#define BN   128   // feature-cols per block
#define KC   32    // K staged per LDS round
#define PAD  36    // LDS row stride in floats (mult of 4 -> 16B-aligned b128; even -> b64 frag loads)

__device__ __forceinline__ float softplus_f(float v) {
    return fmaxf(v, 0.f) + log1pf(expf(-fabsf(v)));
}

// CDNA5 async global->LDS copy (ASYNCcnt-tracked, GV addressing mode).
// lds_off: LDS byte address (low 32 bits of generic shared pointer);
// gaddr:   per-lane 64-bit global address (VGPR pair).
__device__ __forceinline__ void async_copy_b128(unsigned lds_off, const float* gaddr) {
    asm volatile("global_load_async_to_lds_b128 %0, %1, off"
                 :: "v"(lds_off), "v"(gaddr)
                 : "memory");
}
__device__ __forceinline__ void wait_async0() {
    asm volatile("s_wait_asynccnt 0" ::: "memory");
}

// D[n,m] = sum_k A[n,k]*B[m,k] (+ optional second pass A2,B2) with epilogue EPI:
//  EPI 0: D = acc + bias0[m]
//  EPI 1: D = tanh(acc + cmat[n,m] + prev[n]*wcol[m*wstride])   (prev may be null -> 0)
//  EPI 2: D = tanh(acc + bias0[m] + bias1[m])
//  EPI 3: D = relu(acc + bias0[m])
template<int EPI, bool BVEC4>
__global__ __launch_bounds__(256)
void gemm_wmma(const float* __restrict__ A,  int lda,
               const float* __restrict__ B,  int ldb,  int K,
               const float* __restrict__ A2, int lda2,
               const float* __restrict__ B2, int ldb2, int K2,
               float* __restrict__ D, int ldd,
               const float* __restrict__ bias0,
               const float* __restrict__ bias1,
               const float* __restrict__ cmat, int ldc,
               const float* __restrict__ prevvec, int prevStride,
               const float* __restrict__ wcol, int wstride)
{
    __shared__ float As[2][BM * PAD];
    __shared__ float Bs[2][BN * PAD];

    const int tid  = threadIdx.x;
    const int wave = tid >> 5;
    const int lane = tid & 31;
    const int wm   = wave & 3;   // 4 waves along batch (16 rows each)
    const int wn   = wave >> 2;  // 2 waves along features (64 cols each)
    const int nblock = blockIdx.x * BM;
    const int mblock = blockIdx.y * BN;

    v8f acc[4] = {};

    const int half2 = (lane >> 4) * 2;  // lanes 0-15 hold K=0,1; lanes 16-31 hold K=2,3
    const int lr    = lane & 15;

    if constexpr (BVEC4) {
        // ---- async double-buffered pipeline: one barrier per K-chunk ----
        const int nc1 = K / KC;
        const int nc2 = A2 ? (K2 / KC) : 0;
        const int nch = nc1 + nc2;

        auto issue_chunk = [&](int c, int buf) {
            const float* Ap; const float* Bp; int ldA, ldB, k0;
            if (c < nc1) { Ap = A;  Bp = B;  ldA = lda;  ldB = ldb;  k0 = c * KC; }
            else         { Ap = A2; Bp = B2; ldA = lda2; ldB = ldb2; k0 = (c - nc1) * KC; }
            const unsigned aBase = (unsigned)(uintptr_t)&As[buf][0];
            const unsigned bBase = (unsigned)(uintptr_t)&Bs[buf][0];
            #pragma unroll
            for (int i = tid; i < BM * (KC / 4); i += 256) {   // 2 per thread
                const int r = i >> 3, c4 = (i & 7) * 4;
                async_copy_b128(aBase + (unsigned)((r * PAD + c4) * 4),
                                Ap + (size_t)(nblock + r) * ldA + k0 + c4);
            }
            #pragma unroll
            for (int i = tid; i < BN * (KC / 4); i += 256) {   // 4 per thread
                const int r = i >> 3, c4 = (i & 7) * 4;
                async_copy_b128(bBase + (unsigned)((r * PAD + c4) * 4),
                                Bp + (size_t)(mblock + r) * ldB + k0 + c4);
            }
        };

        issue_chunk(0, 0);
        for (int c = 0; c < nch; ++c) {
            const int cur = c & 1;
            wait_async0();          // my chunk-c transfers are in LDS
            __syncthreads();        // everyone's are; prior buffer free for refill
            if (c + 1 < nch) issue_chunk(c + 1, cur ^ 1);

            const float* as = &As[cur][0];
            const float* bs = &Bs[cur][0];
            #pragma unroll
            for (int kk = 0; kk < KC; kk += 4) {
                const v2f a = *(const v2f*)&as[(wm * 16 + lr) * PAD + kk + half2];
                #pragma unroll
                for (int j = 0; j < 4; ++j) {
                    const v2f b = *(const v2f*)&bs[(wn * 64 + j * 16 + lr) * PAD + kk + half2];
                    acc[j] = __builtin_amdgcn_wmma_f32_16x16x4_f32(
                        false, a, false, b, (short)0, acc[j], false, false);
                }
            }
        }
    } else {
        // ---- synchronous scalar-B path (unaligned ldb, e.g. 257; runs once per call) ----
        for (int k0 = 0; k0 < K; k0 += KC) {
            #pragma unroll
            for (int i = tid; i < BM * (KC / 4); i += 256) {
                const int r = i >> 3, c4 = (i & 7) * 4;
                const float4 v = *(const float4*)(A + (size_t)(nblock + r) * lda + k0 + c4);
                *(float4*)&As[0][r * PAD + c4] = v;
            }
            #pragma unroll
            for (int i = tid; i < BN * KC; i += 256) {
                const int r = i >> 5, c = i & 31;
                Bs[0][r * PAD + c] = B[(size_t)(mblock + r) * ldb + k0 + c];
            }
            __syncthreads();
            #pragma unroll
            for (int kk = 0; kk < KC; kk += 4) {
                const v2f a = *(const v2f*)&As[0][(wm * 16 + lr) * PAD + kk + half2];
                #pragma unroll
                for (int j = 0; j < 4; ++j) {
                    const v2f b = *(const v2f*)&Bs[0][(wn * 64 + j * 16 + lr) * PAD + kk + half2];
                    acc[j] = __builtin_amdgcn_wmma_f32_16x16x4_f32(
                        false, a, false, b, (short)0, acc[j], false, false);
                }
            }
            __syncthreads();
        }
    }

    // ---- epilogue ----
    const int mrowoff = (lane >= 16) ? 8 : 0;   // C/D layout: VGPR r holds M=r (lo half) / M=r+8 (hi half)
    #pragma unroll
    for (int j = 0; j < 4; ++j) {
        const int m = mblock + wn * 64 + j * 16 + lr;
        float bv = 0.f;
        if (EPI == 0 || EPI == 2 || EPI == 3) bv = bias0[m];
        if (EPI == 2) bv += bias1[m];
        float wcv = 0.f;
        if (EPI == 1) wcv = wcol[(size_t)m * wstride];
        #pragma unroll
        for (int r = 0; r < 8; ++r) {
            const int n = nblock + wm * 16 + mrowoff + r;
            float d = acc[j][r];
            if (EPI == 0) {
                d += bv;
            } else if (EPI == 1) {
                const float prev = prevvec ? prevvec[(size_t)n * prevStride] : 0.f;
                d = tanhf(d + cmat[(size_t)n * ldc + m] + prev * wcv);
            } else if (EPI == 2) {
                d = tanhf(d + bv);
            } else {
                d = fmaxf(d + bv, 0.f);
            }
            D[(size_t)n * ldd + m] = d;
        }
    }
}

// once-per-call init: x_o = x*mask, c0bias[m] = b_ih0+b_hh0 + sum_d mask[d]*W_ih0[m,129+d],
// zero h0/h1 initial state and p accumulator
__global__ void preamble_kernel(const float* __restrict__ x, const float* __restrict__ mask,
                                const float* __restrict__ W_ih0,
                                const float* __restrict__ b_ih0, const float* __restrict__ b_hh0,
                                float* __restrict__ x_o, float* __restrict__ c0bias,
                                float* __restrict__ h0a, float* __restrict__ h1a,
                                float* __restrict__ p)
{
    const int i = blockIdx.x * blockDim.x + threadIdx.x;
    if (i < NBAT * DIM)  x_o[i] = x[i] * mask[i & (DIM - 1)];
    if (i < NBAT * HID) { h0a[i] = 0.f; h1a[i] = 0.f; }
    if (i < NBAT)        p[i] = 0.f;
    if (i < HID) {
        float s = b_ih0[i] + b_hh0[i];
        #pragma unroll 4
        for (int d = 0; d < DIM; ++d)
            s += mask[d] * W_ih0[(size_t)i * IN_W + 1 + DIM + d];
        c0bias[i] = s;
    }
}

// per-step GMM log-prob: one wave per batch row, 4 components per lane, shfl-xor reduce
__global__ __launch_bounds__(256)
void logprob_kernel(const float* __restrict__ mc, const float* __restrict__ x,
                    int t, float* __restrict__ p)
{
    const int wave = threadIdx.x >> 5;
    const int lane = threadIdx.x & 31;
    const int n = blockIdx.x * 8 + wave;
    const float xt = x[(size_t)n * DIM + t];

    const float4 mean = *(const float4*)(mc + (size_t)n * (2 * NC) + lane * 4);
    const float4 cov  = *(const float4*)(mc + (size_t)n * (2 * NC) + NC + lane * 4);

    float quad = 0.f, lsum = 0.f;
    {
        const float mv[4] = {mean.x, mean.y, mean.z, mean.w};
        const float cv[4] = {cov.x,  cov.y,  cov.z,  cov.w};
        #pragma unroll
        for (int i = 0; i < 4; ++i) {
            const float sp = softplus_f(cv[i]);   // 1/prec^2 == softplus(cov)
            const float dm = xt - mv[i];
            quad += dm * dm * sp;
            lsum += logf(sp);                      // log(prec) = -0.5*log(sp)
        }
    }
    #pragma unroll
    for (int off = 16; off; off >>= 1) {
        quad += __shfl_xor(quad, off);
        lsum += __shfl_xor(lsum, off);
    }
    if (lane == 0) {
        const float LOG2PI = 1.8378770664093453f;
        p[n] += -0.5f * ((float)DIM * LOG2PI + quad) - 0.5f * lsum;
    }
}

// out = [h1_final (N*HID) | p (N)]
__global__ void finalize_kernel(const float* __restrict__ h1fin, const float* __restrict__ p,
                                float* __restrict__ out)
{
    const int i = blockIdx.x * blockDim.x + threadIdx.x;
    if (i < NBAT * HID)             out[i] = h1fin[i];
    else if (i < NBAT * HID + NBAT) out[i] = p[i - NBAT * HID];
}

extern "C" void kernel_launch(void* const* d_in, const int* in_sizes, int n_in,
                              void* d_out, int out_size, void* d_ws, size_t ws_size,
                              hipStream_t stream)
{
    const float* x      = (const float*)d_in[0];
    const float* mask   = (const float*)d_in[1];
    const float* W_ih0  = (const float*)d_in[2];
    const float* W_hh0  = (const float*)d_in[3];
    const float* b_ih0  = (const float*)d_in[4];
    const float* b_hh0  = (const float*)d_in[5];
    const float* W_ih1  = (const float*)d_in[6];
    const float* W_hh1  = (const float*)d_in[7];
    const float* b_ih1  = (const float*)d_in[8];
    const float* b_hh1  = (const float*)d_in[9];
    const float* W1     = (const float*)d_in[10];
    const float* b1     = (const float*)d_in[11];
    const float* W2     = (const float*)d_in[12];
    const float* b2     = (const float*)d_in[13];

    float* ws     = (float*)d_ws;
    float* x_o    = ws;  ws += NBAT * DIM;   // 65536
    float* c0bias = ws;  ws += HID;          // 512
    float* c0     = ws;  ws += NBAT * HID;   // 262144
    float* h0buf0 = ws;  ws += NBAT * HID;
    float* h0buf1 = ws;  ws += NBAT * HID;
    float* h1buf0 = ws;  ws += NBAT * HID;
    float* h1buf1 = ws;  ws += NBAT * HID;
    float* hid    = ws;  ws += NBAT * MLP_H; // 1048576
    float* mc     = ws;  ws += NBAT * 2 * NC;// 131072
    float* p      = ws;  ws += NBAT;

    float* h0b[2] = {h0buf0, h0buf1};
    float* h1b[2] = {h1buf0, h1buf1};

    const dim3 blk(256);

    preamble_kernel<<<dim3((NBAT * HID) / 256), blk, 0, stream>>>(
        x, mask, W_ih0, b_ih0, b_hh0, x_o, c0bias, h0b[0], h1b[0], p);

    // c0 = x_o @ W_ih0[:,1:129]^T + c0bias  (B row stride 257 -> scalar B path)
    gemm_wmma<0, false><<<dim3(NBAT / BM, HID / BN), blk, 0, stream>>>(
        x_o, DIM, W_ih0 + 1, IN_W, DIM,
        nullptr, 0, nullptr, 0, 0,
        c0, HID, c0bias, nullptr, nullptr, 0, nullptr, 0, nullptr, 0);

    for (int t = 0; t < DIM; ++t) {
        const int rp = t & 1;
        const int wp = rp ^ 1;
        const float* prevcol = (t == 0) ? nullptr : (x + (t - 1)); // stride DIM

        // h0_new = tanh(c0 + prev*W_ih0[:,0] + h0 @ W_hh0^T)
        gemm_wmma<1, true><<<dim3(NBAT / BM, HID / BN), blk, 0, stream>>>(
            h0b[rp], HID, W_hh0, HID, HID,
            nullptr, 0, nullptr, 0, 0,
            h0b[wp], HID, nullptr, nullptr, c0, HID, prevcol, DIM, W_ih0, IN_W);

        // h1_new = tanh(h0_new @ W_ih1^T + h1 @ W_hh1^T + b_ih1 + b_hh1)  (dual-GEMM)
        gemm_wmma<2, true><<<dim3(NBAT / BM, HID / BN), blk, 0, stream>>>(
            h0b[wp], HID, W_ih1, HID, HID,
            h1b[rp], HID, W_hh1, HID, HID,
            h1b[wp], HID, b_ih1, b_hh1, nullptr, 0, nullptr, 0, nullptr, 0);

        // hid = relu(h1_new @ W1^T + b1)
        gemm_wmma<3, true><<<dim3(NBAT / BM, MLP_H / BN), blk, 0, stream>>>(
            h1b[wp], HID, W1, HID, HID,
            nullptr, 0, nullptr, 0, 0,
            hid, MLP_H, b1, nullptr, nullptr, 0, nullptr, 0, nullptr, 0);

        // mc = hid @ W2^T + b2
        gemm_wmma<0, true><<<dim3(NBAT / BM, (2 * NC) / BN), blk, 0, stream>>>(
            hid, MLP_H, W2, MLP_H, MLP_H,
            nullptr, 0, nullptr, 0, 0,
            mc, 2 * NC, b2, nullptr, nullptr, 0, nullptr, 0, nullptr, 0);

        // p += logprob(x[:,t]; mc)
        logprob_kernel<<<dim3(NBAT / 8), blk, 0, stream>>>(mc, x, t, p);
    }

    // after 128 steps (even), final states live in buffer 0
    finalize_kernel<<<dim3((NBAT * HID + NBAT + 255) / 256), blk, 0, stream>>>(
        h1b[0], p, (float*)d_out);
}